// GroupedLinearCF_82643760709938
// MI455X (gfx1250) — compile-verified
//
#include <hip/hip_runtime.h>

// GroupedLinearCF for MI455X (gfx1250, wave32, WMMA).
// x:[8,64,1024,256] f32, weight:[32,8,64,64] f32 (flat: f*4096), bias:[64,1,256] f32
// out:[8,64,1024,256] f32.
// Per f: GEMM [8192 x 64] x [64 x 64] via V_WMMA_F32_16X16X4_F32.

typedef __attribute__((ext_vector_type(2))) float v2f;
typedef __attribute__((ext_vector_type(8))) float v8f;

#define B_      8
#define CIN     64
#define T_      1024
#define F_      256
#define COUT    64
#define MTILE   32               // t-rows per workgroup (within one b; 1024 % 32 == 0)
#define FTILE   8                // f-columns per workgroup (one per wave)
#define XSTRIDE 513              // m-stride in floats (64*8 + 1) -> conflict-free A reads
#define YSTRIDE 260              // o-stride in floats (32*8 + 4) -> float4-aligned
#define LDS_FLOATS 16640         // max(32*513, 64*260) rounded up -> 66560 bytes

extern "C" __global__ void __launch_bounds__(256)
glcf_wmma_kernel(const float* __restrict__ x,
                 const float* __restrict__ w,
                 const float* __restrict__ bias,
                 float* __restrict__ out)
{
    extern __shared__ float lds[];

    const int tid   = threadIdx.x;
    const int lane  = tid & 31;
    const int wv    = tid >> 5;          // wave id 0..7 == local f
    const int f_base = blockIdx.x * FTILE;   // 32 f-tiles (fast-varying: share cache lines)
    const int m_base = blockIdx.y * MTILE;   // 256 m-tiles
    const int b      = m_base >> 10;         // / T_
    const int t_base = m_base & (T_ - 1);

    // ---------------- Stage 1: x tile -> LDS, coalesced float4 along f ----------
    // 32 m * 64 i * 8 f floats = 4096 float4; 16 iters * 256 threads.
    #pragma unroll
    for (int it = 0; it < 16; ++it) {
        const int q  = it * 256 + tid;
        const int f4 = q & 1;            // which float4 of the 8-float row
        const int i  = (q >> 1) & 63;    // C_in
        const int m  = q >> 7;           // 0..31
        const float4 v = *(const float4*)(x +
            (((size_t)(b * CIN + i) * T_ + (size_t)(t_base + m)) * F_ + f_base + f4 * 4));
        float* dst = lds + m * XSTRIDE + i * FTILE + f4 * 4;   // 4B-aligned only (odd stride)
        dst[0] = v.x; dst[1] = v.y; dst[2] = v.z; dst[3] = v.w;
    }
    __syncthreads();

    // ---------------- Stage 2: per-wave GEMM 32x64 * 64x64 via WMMA f32 ----------
    const int f  = f_base + wv;
    const float* __restrict__ Wf = w + (size_t)f * (CIN * COUT);  // 64x64, o contiguous
    __builtin_prefetch(Wf, 0, 1);                                  // global_prefetch

    const int hk = lane >> 4;            // 0: lanes 0-15 (K pair 0,1) ; 1: lanes 16-31 (K pair 2,3)
    const int ln = lane & 15;            // A: M row / B: N column / C: N column

    v8f acc[2][4];
    #pragma unroll
    for (int s = 0; s < 2; ++s)
        #pragma unroll
        for (int nt = 0; nt < 4; ++nt)
            #pragma unroll
            for (int r = 0; r < 8; ++r)
                acc[s][nt][r] = 0.0f;

    // bias fragment: same o per lane for all 8 rows of a C tile
    float bfrag[4];
    #pragma unroll
    for (int nt = 0; nt < 4; ++nt)
        bfrag[nt] = bias[(size_t)(nt * 16 + ln) * F_ + f];

    #pragma unroll
    for (int k = 0; k < 16; ++k) {       // K = 64 in steps of 4
        const int i0 = k * 4 + hk * 2;
        // A fragments (16x4 f32): VGPR0 = K{0|2}, VGPR1 = K{1|3}; conflict-free banks
        v2f a0, a1;
        a0.x = lds[ ln        * XSTRIDE +  i0      * FTILE + wv];
        a0.y = lds[ ln        * XSTRIDE + (i0 + 1) * FTILE + wv];
        a1.x = lds[(ln + 16)  * XSTRIDE +  i0      * FTILE + wv];
        a1.y = lds[(ln + 16)  * XSTRIDE + (i0 + 1) * FTILE + wv];
        #pragma unroll
        for (int nt = 0; nt < 4; ++nt) {
            // B fragment (4x16 f32) straight from L2-hot weight
            v2f bb;
            bb.x = Wf[(size_t) i0      * COUT + nt * 16 + ln];
            bb.y = Wf[(size_t)(i0 + 1) * COUT + nt * 16 + ln];
            acc[0][nt] = __builtin_amdgcn_wmma_f32_16x16x4_f32(
                false, a0, false, bb, (short)0, acc[0][nt], false, false);
            acc[1][nt] = __builtin_amdgcn_wmma_f32_16x16x4_f32(
                false, a1, false, bb, (short)0, acc[1][nt], false, false);
        }
    }

    // bias add
    #pragma unroll
    for (int s = 0; s < 2; ++s)
        #pragma unroll
        for (int nt = 0; nt < 4; ++nt)
            #pragma unroll
            for (int r = 0; r < 8; ++r)
                acc[s][nt][r] += bfrag[nt];

    // ---------------- Stage 3: C tiles -> LDS transpose [o][m][f] ----------------
    __syncthreads();                      // all X reads done; safe to overwrite LDS
    #pragma unroll
    for (int s = 0; s < 2; ++s)
        #pragma unroll
        for (int nt = 0; nt < 4; ++nt) {
            const int o = nt * 16 + ln;
            #pragma unroll
            for (int r = 0; r < 8; ++r) {
                const int m = s * 16 + hk * 8 + r;   // C layout: VGPR r = M{r | r+8}
                lds[o * YSTRIDE + m * FTILE + wv] = acc[s][nt][r];
            }
        }
    __syncthreads();

    // ---------------- Stage 4: coalesced float4 store to out ---------------------
    float* outBase = out + (((size_t)b * COUT) * T_ + t_base) * F_ + f_base;
    #pragma unroll
    for (int it = 0; it < 16; ++it) {
        const int q  = it * 256 + tid;
        const int f4 = q & 1;
        const int m  = (q >> 1) & 31;
        const int o  = q >> 6;
        const float4 v = *(const float4*)(lds + o * YSTRIDE + m * FTILE + f4 * 4);
        *(float4*)(outBase + ((size_t)o * T_ + m) * F_ + f4 * 4) = v;
    }
}

extern "C" void kernel_launch(void* const* d_in, const int* in_sizes, int n_in,
                              void* d_out, int out_size, void* d_ws, size_t ws_size,
                              hipStream_t stream) {
    const float* x    = (const float*)d_in[0];
    const float* wgt  = (const float*)d_in[1];
    const float* bias = (const float*)d_in[2];
    float* out = (float*)d_out;

    const size_t shmem = LDS_FLOATS * sizeof(float);   // 66560 B (> 64KB; WGP allows 320KB)
    (void)hipFuncSetAttribute((const void*)glcf_wmma_kernel,
                              hipFuncAttributeMaxDynamicSharedMemorySize, (int)shmem);

    dim3 grid(F_ / FTILE, (B_ * T_) / MTILE);   // (32, 256)
    dim3 block(256);
    glcf_wmma_kernel<<<grid, block, shmem, stream>>>(x, wgt, bias, out);
}